// CircularConvolution1_10075993276566
// MI455X (gfx1250) — compile-verified
//
#include <hip/hip_runtime.h>
#include <hip/hip_bf16.h>

typedef __attribute__((ext_vector_type(2))) float v2f;
typedef __attribute__((ext_vector_type(4))) float v4f;
typedef __attribute__((ext_vector_type(8))) float v8f;

#define MSIZE 8192
#define MMASK 8191

// ---------------------------------------------------------------------------
// Kernel 1: materialize key_values[j,k] = key[(j-k) mod 8192]   (256 MB, the
// bandwidth-dominant part: ~11 us floor at 23.3 TB/s).
// Stage key doubled in LDS so key_values[j,k] = key2[j - k + 8192] (no mod),
// then stream coalesced 16B non-temporal stores.
// ---------------------------------------------------------------------------
#define KV_ROWS 16
#define KV_THREADS 256

__global__ __launch_bounds__(KV_THREADS)
void kv_gen_kernel(const float* __restrict__ key, float* __restrict__ kv) {
  __shared__ float key2[2 * MSIZE];  // 64 KB LDS
  const int tid = threadIdx.x;
  for (int i = tid; i < MSIZE; i += KV_THREADS) {
    const float v = key[i];
    key2[i] = v;
    key2[i + MSIZE] = v;
  }
  __syncthreads();

  const int j0 = blockIdx.x * KV_ROWS;
  #pragma unroll
  for (int r = 0; r < KV_ROWS; ++r) {
    const int j = j0 + r;
    float* dst = kv + (size_t)j * MSIZE;
    const int base0 = j + MSIZE;  // index for k = 0
    // thread t covers k = 4*t, 4*t + 1024, ... ; 16B stores, coalesced per wave
    for (int c = tid * 4; c < MSIZE; c += KV_THREADS * 4) {
      const int b = base0 - c;  // in [4, 16383]
      v4f v;
      v.x = key2[b];
      v.y = key2[b - 1];
      v.z = key2[b - 2];
      v.w = key2[b - 3];
      __builtin_nontemporal_store(v, (v4f*)(dst + c));
    }
  }
}

// ---------------------------------------------------------------------------
// Kernel 2: result = circulant @ x via V_WMMA_F32_16X16X4_F32 (f32-exact).
// One wave computes 16 output rows. A = 16x4 slice of the circulant gathered
// from LDS; B = 4x16 with x[kk..kk+3] in column N=0 (lanes 0/16), zero else.
// Column 0 of the 16x16 f32 D accumulator holds the 16 dot products.
// ---------------------------------------------------------------------------
#define MV_THREADS 256
#define MV_WAVES (MV_THREADS / 32)

__global__ __launch_bounds__(MV_THREADS)
void mv_wmma_kernel(const float* __restrict__ key, const float* __restrict__ x,
                    float* __restrict__ out) {
  __shared__ float key_s[MSIZE];                  // 32 KB
  __shared__ __align__(16) float x_s[MSIZE];      // 32 KB
  const int tid = threadIdx.x;
  for (int i = tid; i < MSIZE; i += MV_THREADS) {
    key_s[i] = key[i];
    x_s[i] = x[i];
  }
  __syncthreads();

  const int wave = tid >> 5;
  const int lane = tid & 31;
  const int j0 = (blockIdx.x * MV_WAVES + wave) * 16;  // 16 rows per wave

  // A-matrix 16x4 f32 layout: lane l -> M = l%16, K = (l/16)*2 + reg
  const int mrow  = lane & 15;
  const int kbase = (lane >> 4) * 2;
  const int basel = j0 + mrow - kbase;   // A[reg0] = key[(basel - kk) & MMASK]
  const bool bsel = (mrow == 0);         // lanes 0 and 16 hold B column N=0

  v8f c0 = {}, c1 = {};
  for (int kk = 0; kk < MSIZE; kk += 8) {
    {
      const int i0 = (basel - kk) & MMASK;
      const int i1 = (i0 - 1) & MMASK;
      v2f a; a.x = key_s[i0]; a.y = key_s[i1];
      const v2f bb = *(const v2f*)&x_s[kk + kbase];   // 8B-aligned
      v2f b; b.x = bsel ? bb.x : 0.0f; b.y = bsel ? bb.y : 0.0f;
      c0 = __builtin_amdgcn_wmma_f32_16x16x4_f32(false, a, false, b,
                                                 (short)0, c0, false, false);
    }
    {
      const int i0 = (basel - kk - 4) & MMASK;
      const int i1 = (i0 - 1) & MMASK;
      v2f a; a.x = key_s[i0]; a.y = key_s[i1];
      const v2f bb = *(const v2f*)&x_s[kk + 4 + kbase];
      v2f b; b.x = bsel ? bb.x : 0.0f; b.y = bsel ? bb.y : 0.0f;
      c1 = __builtin_amdgcn_wmma_f32_16x16x4_f32(false, a, false, b,
                                                 (short)0, c1, false, false);
    }
  }
  const v8f c = c0 + c1;

  // f32 C/D 16x16 layout: column N=0 lives in lane 0 (M=0..7 in regs 0..7)
  // and lane 16 (M=8..15 in regs 0..7).
  if (lane == 0) {
    #pragma unroll
    for (int r = 0; r < 8; ++r) out[j0 + r] = c[r];
  } else if (lane == 16) {
    #pragma unroll
    for (int r = 0; r < 8; ++r) out[j0 + 8 + r] = c[r];
  }
}

// ---------------------------------------------------------------------------
// Launch. d_out = [ result (8192 f32) | key_values (8192*8192 f32) ].
// ---------------------------------------------------------------------------
extern "C" void kernel_launch(void* const* d_in, const int* in_sizes, int n_in,
                              void* d_out, int out_size, void* d_ws, size_t ws_size,
                              hipStream_t stream) {
  (void)in_sizes; (void)n_in; (void)out_size; (void)d_ws; (void)ws_size;
  const float* key = (const float*)d_in[0];   // (8192,)
  const float* x   = (const float*)d_in[1];   // (8192,)
  // d_in[2] = factor: unused by the reference computation.

  float* result = (float*)d_out;
  float* kv     = (float*)d_out + MSIZE;

  // Tiny matvec (runs in a few us, fully f32-exact via WMMA 16x16x4).
  mv_wmma_kernel<<<MSIZE / (16 * MV_WAVES), MV_THREADS, 0, stream>>>(key, x, result);

  // Bandwidth-bound circulant materialization: 512 blocks x 16 rows.
  kv_gen_kernel<<<MSIZE / KV_ROWS, KV_THREADS, 0, stream>>>(key, kv);
}